// CannyEdge_6777458393415
// MI455X (gfx1250) — compile-verified
//
#include <hip/hip_runtime.h>
#include <math.h>

#define IMG_H 2048
#define IMG_W 2048
#define TILE  32
#define GDIM  40   // gray tile  (out + halo 4)
#define BDIM  36   // blurred    (out + halo 2)
#define MDIM  34   // magnitude  (out + halo 1)
#define SG    41   // LDS strides (odd -> no bank conflicts)
#define ST    37
#define SB    37
#define SM    35

static __device__ __forceinline__ int reflectIdx(int i, int n) {
  i = (i < 0) ? -i : i;                 // jnp.pad 'reflect' (edge not repeated)
  i = (i >= n) ? (2 * (n - 1) - i) : i;
  return i;
}

__global__ __launch_bounds__(256) void canny_fused_kernel(
    const float* __restrict__ img, float* __restrict__ out) {
  // LDS plan (floats), stages separated by barriers so regions alias safely:
  //   [0,    4800) : 3x 40x40 raw channel tiles  -> later tmpH[0,1480) / blur[1480,2812) / mg[2812,4002)
  //   [4800, 6440) : gray 40 x SG
  __shared__ float smem[6440];
  float* chan = smem;
  float* gray = smem + 4800;
  float* tmpH = smem;
  float* blur = smem + 1480;
  float* mg   = smem + 2812;

  const int tid = threadIdx.x;
  const int ty0 = blockIdx.y * TILE;
  const int tx0 = blockIdx.x * TILE;
  const size_t NPIX = (size_t)IMG_H * IMG_W;

  // ---- Stage A: CDNA5 async global->LDS copy of the 3 channel tiles (reflect-mapped).
  // Per-lane LDS dest + per-lane global addr; tracked on ASYNCcnt (no VGPR staging).
  for (int idx = tid; idx < GDIM * GDIM; idx += 256) {
    const int r = idx / GDIM, c = idx - r * GDIM;
    const int iy = reflectIdx(ty0 - 4 + r, IMG_H);
    const int ix = reflectIdx(tx0 - 4 + c, IMG_W);
    const size_t goff = (size_t)iy * IMG_W + ix;
#pragma unroll
    for (int ch = 0; ch < 3; ++ch) {
      const unsigned long long gaddr =
          (unsigned long long)(size_t)(img + (size_t)ch * NPIX + goff);
      const unsigned lds = (unsigned)(size_t)(&chan[ch * 1600 + idx]);
      asm volatile("global_load_async_to_lds_b32 %0, %1, off"
                   :: "v"(lds), "v"(gaddr) : "memory");
    }
  }
  asm volatile("s_wait_asynccnt 0" ::: "memory");
  __syncthreads();

  // ---- Stage B: grayscale = 0.299R + 0.587G + 0.114B
  for (int idx = tid; idx < GDIM * GDIM; idx += 256) {
    const int r = idx / GDIM, c = idx - r * GDIM;
    gray[r * SG + c] = 0.299f * chan[idx] + 0.587f * chan[1600 + idx] +
                       0.114f * chan[3200 + idx];
  }
  __syncthreads();

  const float GW0 = 0.40261996f;  // gauss1d(5, sigma=1) normalized
  const float GW1 = 0.24420135f;
  const float GW2 = 0.05448869f;

  // ---- Stage C: horizontal Gaussian (40 rows x 36 cols), overwrites chan
  for (int idx = tid; idx < GDIM * BDIM; idx += 256) {
    const int r = idx / BDIM, c = idx - r * BDIM;
    const float* g = &gray[r * SG + c];
    tmpH[r * ST + c] = GW2 * (g[0] + g[4]) + GW1 * (g[1] + g[3]) + GW0 * g[2];
  }
  __syncthreads();

  // ---- Stage D: vertical Gaussian (36 x 36)
  for (int idx = tid; idx < BDIM * BDIM; idx += 256) {
    const int r = idx / BDIM, c = idx - r * BDIM;
    const float* t = &tmpH[r * ST + c];
    blur[r * SB + c] =
        GW2 * (t[0] + t[4 * ST]) + GW1 * (t[ST] + t[3 * ST]) + GW0 * t[2 * ST];
  }
  __syncthreads();

  // ---- Stage E: Sobel + magnitude on 34x34 (zero outside image = NMS zero-pad)
  for (int idx = tid; idx < MDIM * MDIM; idx += 256) {
    const int r = idx / MDIM, c = idx - r * MDIM;
    const int py = ty0 - 1 + r, px = tx0 - 1 + c;
    float m = 0.0f;
    if (py >= 0 && py < IMG_H && px >= 0 && px < IMG_W) {
      // clamp-to-edge (reference pads blurred with 'edge'), map into LDS tile
      const int rup = max(py - 1, 0)         - (ty0 - 2);
      const int r0  = py                     - (ty0 - 2);
      const int rdn = min(py + 1, IMG_H - 1) - (ty0 - 2);
      const int clf = max(px - 1, 0)         - (tx0 - 2);
      const int c0  = px                     - (tx0 - 2);
      const int crt = min(px + 1, IMG_W - 1) - (tx0 - 2);
      const float bul = blur[rup*SB+clf], bum = blur[rup*SB+c0], bur = blur[rup*SB+crt];
      const float bml = blur[r0 *SB+clf],                         bmr = blur[r0 *SB+crt];
      const float bdl = blur[rdn*SB+clf], bdm = blur[rdn*SB+c0], bdr = blur[rdn*SB+crt];
      const float sgx = (bur + 2.0f*bmr + bdr) - (bul + 2.0f*bml + bdl);
      const float sgy = (bdl + 2.0f*bdm + bdr) - (bul + 2.0f*bum + bur);
      m = sqrtf(sgx * sgx + sgy * sgy + 1e-6f);
    }
    mg[r * SM + c] = m;
  }
  __syncthreads();

  // ---- Stage F: direction + NMS + tiled 3-channel output (float4 stores)
  // nibble-packed (dy+1, dx+1) for _OFFS table, avoids scratch-indexed arrays
  const unsigned DYP = 0x00012221u;
  const unsigned DXP = 0x21000122u;
  {
    const int r  = tid >> 3;        // 0..31
    const int c4 = (tid & 7) * 4;   // 0,4,...,28
    const int py = ty0 + r;
    float4 res;
    float* resp = &res.x;
#pragma unroll
    for (int k = 0; k < 4; ++k) {
      const int c = c4 + k;
      const int px = tx0 + c;
      // recompute gx,gy from LDS blur for the direction index
      const int rup = max(py - 1, 0)         - (ty0 - 2);
      const int r0  = py                     - (ty0 - 2);
      const int rdn = min(py + 1, IMG_H - 1) - (ty0 - 2);
      const int clf = max(px - 1, 0)         - (tx0 - 2);
      const int c0  = px                     - (tx0 - 2);
      const int crt = min(px + 1, IMG_W - 1) - (tx0 - 2);
      const float bul = blur[rup*SB+clf], bum = blur[rup*SB+c0], bur = blur[rup*SB+crt];
      const float bml = blur[r0 *SB+clf],                         bmr = blur[r0 *SB+crt];
      const float bdl = blur[rdn*SB+clf], bdm = blur[rdn*SB+c0], bdr = blur[rdn*SB+crt];
      const float sgx = (bur + 2.0f*bmr + bdr) - (bul + 2.0f*bml + bdl);
      const float sgy = (bdl + 2.0f*bdm + bdr) - (bul + 2.0f*bum + bur);

      const float m = mg[(r + 1) * SM + (c + 1)];
      // round(rad2deg(atan2)/45) == rint(atan2 * 4/pi); rintf = half-even = jnp.round
      const int o   = (int)rintf(atan2f(sgy, sgx) * 1.2732395447351628f);
      const int pos = (o + 8) & 7;
      const int neg = (pos + 4) & 7;
      const int dyp = (int)((DYP >> (pos * 4)) & 0xF) - 1;
      const int dxp = (int)((DXP >> (pos * 4)) & 0xF) - 1;
      const int dyn = (int)((DYP >> (neg * 4)) & 0xF) - 1;
      const int dxn = (int)((DXP >> (neg * 4)) & 0xF) - 1;
      const float cp = m - mg[(r + 1 + dyp) * SM + (c + 1 + dxp)];
      const float cn = m - mg[(r + 1 + dyn) * SM + (c + 1 + dxn)];
      resp[k] = (fminf(cp, cn) > 0.0f) ? m : 0.0f;
    }
    const size_t o0 = (size_t)py * IMG_W + (size_t)(tx0 + c4);
    *reinterpret_cast<float4*>(out + o0)            = res;
    *reinterpret_cast<float4*>(out + o0 + NPIX)     = res;
    *reinterpret_cast<float4*>(out + o0 + 2 * NPIX) = res;
  }
}

extern "C" void kernel_launch(void* const* d_in, const int* in_sizes, int n_in,
                              void* d_out, int out_size, void* d_ws, size_t ws_size,
                              hipStream_t stream) {
  (void)in_sizes; (void)n_in; (void)d_ws; (void)ws_size; (void)out_size;
  const float* img = (const float*)d_in[0];   // (3, 2048, 2048) fp32
  float* out = (float*)d_out;                 // (3, 2048, 2048) fp32
  dim3 grid(IMG_W / TILE, IMG_H / TILE);      // 64 x 64 tiles
  canny_fused_kernel<<<grid, 256, 0, stream>>>(img, out);
}